// DiffWire_54142357733572
// MI455X (gfx1250) — compile-verified
//
#include <hip/hip_runtime.h>

#define DEV __device__ __forceinline__

typedef __attribute__((ext_vector_type(2))) float v2f;
typedef __attribute__((ext_vector_type(8))) float v8f;

constexpr int Bn  = 16;
constexpr int Nn  = 1024;
constexpr int INF = 128;
constexpr int Hf  = 32;
constexpr int KMC = 16;
constexpr int NT  = Nn / 16;
constexpr float EPSF = 1e-15f;

DEV v8f wmma4(v2f a, v2f b, v8f c) {
  // D = A(16x4,f32) * B(4x16,f32) + C(16x16,f32)  -> v_wmma_f32_16x16x4_f32
  return __builtin_amdgcn_wmma_f32_16x16x4_f32(false, a, false, b, (short)0, c,
                                               false, false);
}
DEV v8f vzero8() {
  v8f v;
  for (int i = 0; i < 8; i++) v[i] = 0.f;
  return v;
}
DEV float block_reduce(float v, float* buf) {
  int tid = threadIdx.x;
  buf[tid] = v;
  __syncthreads();
  for (int s = 128; s > 0; s >>= 1) {
    if (tid < s && tid + s < 256) buf[tid] += buf[tid + s];
    __syncthreads();
  }
  float r = buf[0];
  __syncthreads();
  return r;
}

// ---------------- K1: x1 = x @ W1 + b1 (WMMA, M=16 tile, K=128) ----------------
__global__ __launch_bounds__(32) void k_lin1(const float* __restrict__ x,
                                             const float* __restrict__ W1,
                                             const float* __restrict__ b1,
                                             float* __restrict__ x1) {
  int wg = blockIdx.x, b = wg >> 6, nt = wg & 63;
  int lane = threadIdx.x, lo = lane & 15, hi = lane >> 4;
  const float* xb = x + ((size_t)b * Nn + nt * 16) * INF;
  v8f c0, c1;
  float bias0 = b1[lo], bias1 = b1[16 + lo];
  for (int j = 0; j < 8; j++) { c0[j] = bias0; c1[j] = bias1; }
  for (int k0 = 0; k0 < INF; k0 += 4) {
    v2f a;
    a[0] = xb[lo * INF + k0 + hi * 2 + 0];
    a[1] = xb[lo * INF + k0 + hi * 2 + 1];
    v2f w0, w1;
    w0[0] = W1[(k0 + hi * 2 + 0) * Hf + lo];
    w0[1] = W1[(k0 + hi * 2 + 1) * Hf + lo];
    w1[0] = W1[(k0 + hi * 2 + 0) * Hf + 16 + lo];
    w1[1] = W1[(k0 + hi * 2 + 1) * Hf + 16 + lo];
    c0 = wmma4(a, w0, c0);
    c1 = wmma4(a, w1, c1);
  }
  float* o = x1 + ((size_t)b * Nn + nt * 16) * Hf;
  for (int j = 0; j < 8; j++) {
    int r = j + 8 * hi;
    o[r * Hf + lo] = c0[j];
    o[r * Hf + 16 + lo] = c1[j];
  }
}

// ------------- K2: per-node assignments: s_rw (k=2), s_ct (k=32), Bc, sq -------------
__global__ void k_assign(const float* __restrict__ x1, const float* __restrict__ Wrw,
                         const float* __restrict__ brw, const float* __restrict__ Wct,
                         const float* __restrict__ bct, float* __restrict__ Bc,
                         float* __restrict__ sq) {
  size_t i = (size_t)blockIdx.x * blockDim.x + threadIdx.x;
  if (i >= (size_t)Bn * Nn) return;
  const float* xr = x1 + i * Hf;
  float xv[Hf];
  for (int k = 0; k < Hf; k++) xv[k] = xr[k];
  float l0 = brw[0], l1 = brw[1];
  for (int k = 0; k < Hf; k++) { l0 += xv[k] * Wrw[k * 2 + 0]; l1 += xv[k] * Wrw[k * 2 + 1]; }
  float mx = fmaxf(l0, l1);
  float e0 = __expf(l0 - mx), e1 = __expf(l1 - mx);
  float inv = 1.f / (e0 + e1);
  float s0 = e0 * inv, s1 = e1 * inv;
  float lc[32], m2 = -1e30f;
  for (int c = 0; c < 32; c++) {
    float acc = bct[c];
    for (int k = 0; k < Hf; k++) acc += xv[k] * Wct[k * 32 + c];
    lc[c] = acc;
    m2 = fmaxf(m2, acc);
  }
  float ssum = 0.f;
  for (int c = 0; c < 32; c++) { lc[c] = __expf(lc[c] - m2); ssum += lc[c]; }
  float invs = 1.f / ssum, sqv = 0.f;
  float* row = Bc + i * 48;
  for (int c = 0; c < 32; c++) { float v = lc[c] * invs; row[c] = v; sqv += v * v; }
  row[32] = s0; row[33] = s1; row[34] = 1.0f;
  for (int c = 35; c < 48; c++) row[c] = 0.f;
  sq[i] = sqv;
}

// ------------- K3: Y1 = adj @ Bc  (cols: 0..31 y_ct, 32..33 y_rw, 34 degree) -------------
__global__ __launch_bounds__(32) void k_pass1(const float* __restrict__ adj,
                                              const float* __restrict__ Bc,
                                              float* __restrict__ Y1) {
  int wg = blockIdx.x, b = wg >> 6, nt = wg & 63;
  int lane = threadIdx.x, lo = lane & 15, hi = lane >> 4;
  const float* arow = adj + ((size_t)b * Nn + nt * 16) * Nn;
  const float* bb = Bc + (size_t)b * Nn * 48;
  v8f c0 = vzero8(), c1 = vzero8(), c2 = vzero8();
  for (int m0 = 0; m0 < Nn; m0 += 4) {
    v2f a;
    a[0] = arow[(size_t)lo * Nn + m0 + hi * 2 + 0];
    a[1] = arow[(size_t)lo * Nn + m0 + hi * 2 + 1];
    const float* br = bb + (size_t)(m0 + hi * 2) * 48;
    v2f b0, b1, b2;
    b0[0] = br[lo];      b0[1] = br[48 + lo];
    b1[0] = br[16 + lo]; b1[1] = br[48 + 16 + lo];
    b2[0] = br[32 + lo]; b2[1] = br[48 + 32 + lo];
    c0 = wmma4(a, b0, c0);
    c1 = wmma4(a, b1, c1);
    c2 = wmma4(a, b2, c2);
  }
  float* o = Y1 + ((size_t)b * Nn + nt * 16) * 48;
  for (int j = 0; j < 8; j++) {
    int r = j + 8 * hi;
    o[r * 48 + lo] = c0[j];
    o[r * 48 + 16 + lo] = c1[j];
    o[r * 48 + 32 + lo] = c2[j];
  }
}

// ------------- K4a: per-batch scalars -------------
__global__ void k_reduce1(const float* __restrict__ Bc, const float* __restrict__ Y1,
                          const float* __restrict__ sq, float* __restrict__ scal) {
  __shared__ float buf[256];
  int b = blockIdx.x, tid = threadIdx.x;
  float a_num = 0, a_den = 0, a_sas = 0, a_sds = 0, a_dct = 0, a_vol = 0;
  float a_s00 = 0, a_s01 = 0, a_s11 = 0;
  for (int n = tid; n < Nn; n += 256) {
    const float* bcr = Bc + ((size_t)b * Nn + n) * 48;
    const float* yr = Y1 + ((size_t)b * Nn + n) * 48;
    float s0 = bcr[32], s1 = bcr[33];
    float y0 = yr[32], y1 = yr[33], d = yr[34];
    a_num += s0 * y0 + s1 * y1;
    a_den += (s0 * s0 + s1 * s1) * d;
    float dot = 0.f;
    for (int c = 0; c < 32; c++) dot += bcr[c] * yr[c];
    a_sas += dot;
    float sv = sq[(size_t)b * Nn + n];
    a_sds += d * sv; a_dct += sv; a_vol += d;
    a_s00 += s0 * s0; a_s01 += s0 * s1; a_s11 += s1 * s1;
  }
  float r;
  r = block_reduce(a_num, buf); if (tid == 0) scal[0 * 16 + b] = r;
  r = block_reduce(a_den, buf); if (tid == 0) scal[1 * 16 + b] = r;
  r = block_reduce(a_sas, buf); if (tid == 0) scal[2 * 16 + b] = r;
  r = block_reduce(a_sds, buf); if (tid == 0) scal[3 * 16 + b] = r;
  r = block_reduce(a_dct, buf); if (tid == 0) scal[4 * 16 + b] = r;
  r = block_reduce(a_vol, buf); if (tid == 0) scal[5 * 16 + b] = r;
  r = block_reduce(a_s00, buf); if (tid == 0) scal[6 * 16 + b] = r;
  r = block_reduce(a_s01, buf); if (tid == 0) scal[7 * 16 + b] = r;
  r = block_reduce(a_s11, buf); if (tid == 0) scal[8 * 16 + b] = r;
}

// ------------- K4b: ss_ct[b] = s_ct^T s_ct (32x32) -------------
__global__ void k_ssct(const float* __restrict__ Bc, float* __restrict__ ss) {
  __shared__ float ls[64 * 32];
  int b = blockIdx.x, tid = threadIdx.x;
  float acc[4] = {0, 0, 0, 0};
  int p0 = tid * 4;
  for (int c0 = 0; c0 < Nn; c0 += 64) {
    for (int e = tid; e < 2048; e += 256) {
      int r = e >> 5, cc = e & 31;
      ls[e] = Bc[((size_t)b * Nn + c0 + r) * 48 + cc];
    }
    __syncthreads();
    for (int r = 0; r < 64; r++) {
      const float* row = ls + r * 32;
      for (int q = 0; q < 4; q++) {
        int p = p0 + q;
        acc[q] += row[p >> 5] * row[p & 31];
      }
    }
    __syncthreads();
  }
  for (int q = 0; q < 4; q++) ss[(size_t)b * 1024 + p0 + q] = acc[q];
}

// ------------- K5: fused rewiring pass: AXg = adj_gap@x1, AXc = adj_ct@x1 -------------
__global__ __launch_bounds__(32) void k_pass2(const float* __restrict__ adj,
                                              const float* __restrict__ Bc,
                                              const float* __restrict__ sq,
                                              const float* __restrict__ x1,
                                              const float* __restrict__ volp,
                                              float* __restrict__ AXg,
                                              float* __restrict__ AXc) {
  __shared__ float lgap[256], lct[256];
  int wg = blockIdx.x, b = wg >> 6, nt = wg & 63;
  int lane = threadIdx.x, lo = lane & 15, hi = lane >> 4;
  int n0 = nt * 16;
  const float* bcB = Bc + (size_t)b * Nn * 48;
  const float* sqB = sq + (size_t)b * Nn;
  const float* adjB = adj + (size_t)b * Nn * Nn;
  const float* x1B = x1 + (size_t)b * Nn * Hf;
  float inv_vol = 1.f / (volp[b] + EPSF);
  v2f as[8];
  for (int kk = 0; kk < 8; kk++) {
    const float* r = bcB + (size_t)(n0 + lo) * 48;
    as[kk][0] = r[kk * 4 + hi * 2];
    as[kk][1] = r[kk * 4 + hi * 2 + 1];
  }
  float f_nv[8], sq_nv[8];
  for (int j = 0; j < 8; j++) {
    int r = n0 + j + 8 * hi;
    f_nv[j] = bcB[(size_t)r * 48 + 32];
    sq_nv[j] = sqB[r];
  }
  v8f cg0 = vzero8(), cg1 = vzero8(), cc0 = vzero8(), cc1 = vzero8();
  for (int m0 = 0; m0 < Nn; m0 += 16) {
    v8f c2 = vzero8();
    for (int kk = 0; kk < 8; kk++) {
      v2f bv;
      const float* r = bcB + (size_t)(m0 + lo) * 48;
      bv[0] = r[kk * 4 + hi * 2];
      bv[1] = r[kk * 4 + hi * 2 + 1];
      c2 = wmma4(as[kk], bv, c2);
    }
    float f_m = bcB[(size_t)(m0 + lo) * 48 + 32];
    float sq_m = sqB[m0 + lo];
    for (int j = 0; j < 8; j++) {
      int r = j + 8 * hi;
      float av = adjB[(size_t)(n0 + r) * Nn + m0 + lo];
      float df = f_nv[j] - f_m;
      float g = av * (1.f - df * df);
      float d2v = sq_nv[j] + sq_m - 2.f * c2[j];
      float ct = av * sqrtf(fmaxf(d2v, 0.f) + 1e-12f) * inv_vol;
      lgap[r * 16 + lo] = g;
      lct[r * 16 + lo] = ct;
    }
    __syncthreads();
    for (int kk = 0; kk < 4; kk++) {
      v2f ag, ac;
      ag[0] = lgap[lo * 16 + kk * 4 + hi * 2];
      ag[1] = lgap[lo * 16 + kk * 4 + hi * 2 + 1];
      ac[0] = lct[lo * 16 + kk * 4 + hi * 2];
      ac[1] = lct[lo * 16 + kk * 4 + hi * 2 + 1];
      int kr = m0 + kk * 4 + hi * 2;
      v2f bx0, bx1;
      bx0[0] = x1B[(size_t)kr * Hf + lo];
      bx0[1] = x1B[(size_t)(kr + 1) * Hf + lo];
      bx1[0] = x1B[(size_t)kr * Hf + 16 + lo];
      bx1[1] = x1B[(size_t)(kr + 1) * Hf + 16 + lo];
      cg0 = wmma4(ag, bx0, cg0);
      cg1 = wmma4(ag, bx1, cg1);
      cc0 = wmma4(ac, bx0, cc0);
      cc1 = wmma4(ac, bx1, cc1);
    }
    __syncthreads();
  }
  for (int j = 0; j < 8; j++) {
    int r = j + 8 * hi;
    size_t o = ((size_t)b * Nn + n0 + r) * Hf;
    AXg[o + lo] = cg0[j]; AXg[o + 16 + lo] = cg1[j];
    AXc[o + lo] = cc0[j]; AXc[o + 16 + lo] = cc1[j];
  }
}

// ------------- K6: DenseGraphConv epilogue: out = AX@Wrel + brel + x@Wroot -------------
__global__ void k_dgc(const float* __restrict__ AX, const float* __restrict__ xin,
                      const float* __restrict__ Wrel, const float* __restrict__ brel,
                      const float* __restrict__ Wroot, float* __restrict__ out) {
  size_t i = (size_t)blockIdx.x * blockDim.x + threadIdx.x;
  if (i >= (size_t)Bn * Nn * Hf) return;
  size_t node = i / Hf;
  int f = (int)(i % Hf);
  const float* ar = AX + node * Hf;
  const float* xr = xin + node * Hf;
  float acc = brel[f];
  for (int k = 0; k < Hf; k++) acc += ar[k] * Wrel[k * Hf + f] + xr[k] * Wroot[k * Hf + f];
  out[i] = acc;
}

// ------------- K7: mincut assignments s = softmax(x@W + b), k=16 -------------
__global__ void k_softmax16(const float* __restrict__ xin, const float* __restrict__ W,
                            const float* __restrict__ bb, float* __restrict__ s) {
  size_t i = (size_t)blockIdx.x * blockDim.x + threadIdx.x;
  if (i >= (size_t)Bn * Nn) return;
  const float* xr = xin + i * Hf;
  float xv[Hf];
  for (int k = 0; k < Hf; k++) xv[k] = xr[k];
  float lc[KMC], mx = -1e30f;
  for (int c = 0; c < KMC; c++) {
    float acc = bb[c];
    for (int k = 0; k < Hf; k++) acc += xv[k] * W[k * KMC + c];
    lc[c] = acc;
    mx = fmaxf(mx, acc);
  }
  float ssum = 0.f;
  for (int c = 0; c < KMC; c++) { lc[c] = __expf(lc[c] - mx); ssum += lc[c]; }
  float inv = 1.f / ssum;
  for (int c = 0; c < KMC; c++) s[i * KMC + c] = lc[c] * inv;
}

// ------------- K8: fused rewiring pass: AS = adj_*@s_mc, plus rewired degrees -------------
__global__ __launch_bounds__(32) void k_pass3(const float* __restrict__ adj,
                                              const float* __restrict__ Bc,
                                              const float* __restrict__ sq,
                                              const float* __restrict__ smg,
                                              const float* __restrict__ smc,
                                              const float* __restrict__ volp,
                                              float* __restrict__ ASg,
                                              float* __restrict__ ASc,
                                              float* __restrict__ dgr,
                                              float* __restrict__ dcr) {
  __shared__ float lgap[256], lct[256];
  int wg = blockIdx.x, b = wg >> 6, nt = wg & 63;
  int lane = threadIdx.x, lo = lane & 15, hi = lane >> 4;
  int n0 = nt * 16;
  const float* bcB = Bc + (size_t)b * Nn * 48;
  const float* sqB = sq + (size_t)b * Nn;
  const float* adjB = adj + (size_t)b * Nn * Nn;
  const float* smgB = smg + (size_t)b * Nn * KMC;
  const float* smcB = smc + (size_t)b * Nn * KMC;
  float inv_vol = 1.f / (volp[b] + EPSF);
  v2f as[8];
  for (int kk = 0; kk < 8; kk++) {
    const float* r = bcB + (size_t)(n0 + lo) * 48;
    as[kk][0] = r[kk * 4 + hi * 2];
    as[kk][1] = r[kk * 4 + hi * 2 + 1];
  }
  float f_nv[8], sq_nv[8], rg[8], rc[8];
  for (int j = 0; j < 8; j++) {
    int r = n0 + j + 8 * hi;
    f_nv[j] = bcB[(size_t)r * 48 + 32];
    sq_nv[j] = sqB[r];
    rg[j] = 0.f; rc[j] = 0.f;
  }
  v8f cgs = vzero8(), ccs = vzero8();
  for (int m0 = 0; m0 < Nn; m0 += 16) {
    v8f c2 = vzero8();
    for (int kk = 0; kk < 8; kk++) {
      v2f bv;
      const float* r = bcB + (size_t)(m0 + lo) * 48;
      bv[0] = r[kk * 4 + hi * 2];
      bv[1] = r[kk * 4 + hi * 2 + 1];
      c2 = wmma4(as[kk], bv, c2);
    }
    float f_m = bcB[(size_t)(m0 + lo) * 48 + 32];
    float sq_m = sqB[m0 + lo];
    for (int j = 0; j < 8; j++) {
      int r = j + 8 * hi;
      float av = adjB[(size_t)(n0 + r) * Nn + m0 + lo];
      float df = f_nv[j] - f_m;
      float g = av * (1.f - df * df);
      float d2v = sq_nv[j] + sq_m - 2.f * c2[j];
      float ct = av * sqrtf(fmaxf(d2v, 0.f) + 1e-12f) * inv_vol;
      rg[j] += g; rc[j] += ct;
      lgap[r * 16 + lo] = g;
      lct[r * 16 + lo] = ct;
    }
    __syncthreads();
    for (int kk = 0; kk < 4; kk++) {
      v2f ag, ac;
      ag[0] = lgap[lo * 16 + kk * 4 + hi * 2];
      ag[1] = lgap[lo * 16 + kk * 4 + hi * 2 + 1];
      ac[0] = lct[lo * 16 + kk * 4 + hi * 2];
      ac[1] = lct[lo * 16 + kk * 4 + hi * 2 + 1];
      int kr = m0 + kk * 4 + hi * 2;
      v2f bg, bc2;
      bg[0] = smgB[(size_t)kr * KMC + lo];
      bg[1] = smgB[(size_t)(kr + 1) * KMC + lo];
      bc2[0] = smcB[(size_t)kr * KMC + lo];
      bc2[1] = smcB[(size_t)(kr + 1) * KMC + lo];
      cgs = wmma4(ag, bg, cgs);
      ccs = wmma4(ac, bc2, ccs);
    }
    __syncthreads();
  }
  for (int j = 0; j < 8; j++) {
    int r = j + 8 * hi;
    ASg[((size_t)b * Nn + n0 + r) * KMC + lo] = cgs[j];
    ASc[((size_t)b * Nn + n0 + r) * KMC + lo] = ccs[j];
  }
  __syncthreads();
  for (int j = 0; j < 8; j++) {
    lgap[(j + 8 * hi) * 16 + lo] = rg[j];
    lct[(j + 8 * hi) * 16 + lo] = rc[j];
  }
  __syncthreads();
  if (lane < 16) {
    float sg = 0, sc = 0;
    for (int t = 0; t < 16; t++) { sg += lgap[lane * 16 + t]; sc += lct[lane * 16 + t]; }
    dgr[(size_t)b * Nn + n0 + lane] = sg;
    dcr[(size_t)b * Nn + n0 + lane] = sc;
  }
}

// ------------- K9: pool reductions: out_adj = s^T AS, ss = s^T s, feat = s^T x, den -------------
__global__ void k_pool_reduce(const float* __restrict__ s, const float* __restrict__ AS,
                              const float* __restrict__ xc, const float* __restrict__ drow,
                              float* __restrict__ out_adj, float* __restrict__ ss,
                              float* __restrict__ feat, float* __restrict__ den) {
  __shared__ float lsS[64 * 16], lsA[64 * 16], lsX[64 * 32], lsD[64], buf[256];
  int b = blockIdx.x, tid = threadIdx.x;
  int k_oa = tid >> 4, l_oa = tid & 15;
  int k0 = tid >> 5, f0 = tid & 31;
  int e1 = tid + 256, k1 = e1 >> 5, f1 = e1 & 31;
  float oa = 0, sv = 0, fa = 0, fb = 0, dp = 0;
  for (int c0 = 0; c0 < Nn; c0 += 64) {
    size_t base = (size_t)b * Nn + c0;
    for (int e = tid; e < 1024; e += 256) {
      int r = e >> 4, cc = e & 15;
      lsS[e] = s[(base + r) * 16 + cc];
      lsA[e] = AS[(base + r) * 16 + cc];
    }
    for (int e = tid; e < 2048; e += 256) {
      int r = e >> 5, cc = e & 31;
      lsX[e] = xc[(base + r) * 32 + cc];
    }
    if (tid < 64) lsD[tid] = drow[base + tid];
    __syncthreads();
    for (int r = 0; r < 64; r++) {
      float sk = lsS[r * 16 + k_oa];
      oa += sk * lsA[r * 16 + l_oa];
      sv += sk * lsS[r * 16 + l_oa];
      fa += lsS[r * 16 + k0] * lsX[r * 32 + f0];
      fb += lsS[r * 16 + k1] * lsX[r * 32 + f1];
    }
    if (tid < 64) {
      float s2 = 0;
      const float* row = lsS + tid * 16;
      for (int k = 0; k < 16; k++) s2 += row[k] * row[k];
      dp += lsD[tid] * s2;
    }
    __syncthreads();
  }
  out_adj[b * 256 + tid] = oa;
  ss[b * 256 + tid] = sv;
  feat[b * 512 + tid] = fa;
  feat[b * 512 + tid + 256] = fb;
  float r = block_reduce(dp, buf);
  if (tid == 0) den[b] = r;
}

// ------------- K10: pool finalize (trace, diag-zero, sym-normalize) + second conv -------------
__global__ void k_pool_fin(const float* __restrict__ out_adj_in,
                           const float* __restrict__ featin,
                           const float* __restrict__ Wrel, const float* __restrict__ brel,
                           const float* __restrict__ Wroot, float* __restrict__ x2,
                           float* __restrict__ num_out) {
  __shared__ float oa[256], dk[16], tmat[512], lf[512];
  int b = blockIdx.x, tid = threadIdx.x;
  oa[tid] = out_adj_in[b * 256 + tid];
  lf[tid] = featin[b * 512 + tid];
  lf[tid + 256] = featin[b * 512 + tid + 256];
  __syncthreads();
  if (tid == 0) {
    float tr = 0;
    for (int k = 0; k < 16; k++) tr += oa[k * 16 + k];
    num_out[b] = tr;
  }
  __syncthreads();
  if ((tid >> 4) == (tid & 15)) oa[tid] = 0.f;
  __syncthreads();
  if (tid < 16) {
    float su = 0;
    for (int l = 0; l < 16; l++) su += oa[tid * 16 + l];
    dk[tid] = sqrtf(su) + EPSF;
  }
  __syncthreads();
  for (int e = tid; e < 512; e += 256) {
    int k = e >> 5, f = e & 31;
    float acc = 0;
    for (int l = 0; l < 16; l++) acc += oa[k * 16 + l] / (dk[k] * dk[l]) * lf[l * 32 + f];
    tmat[e] = acc;
  }
  __syncthreads();
  for (int e = tid; e < 512; e += 256) {
    int k = e >> 5, f = e & 31;
    float acc = brel[f];
    for (int j = 0; j < 32; j++)
      acc += tmat[k * 32 + j] * Wrel[j * 32 + f] + lf[k * 32 + j] * Wroot[j * 32 + f];
    x2[(size_t)b * 512 + e] = acc;
  }
}

// ------------- K11: head: concat -> relu MLP -> sum -> relu -> logits -> log_softmax -------------
__global__ void k_head(const float* __restrict__ x2g, const float* __restrict__ x2c,
                       const float* __restrict__ Wcat, const float* __restrict__ bcat,
                       const float* __restrict__ W2, const float* __restrict__ b2,
                       const float* __restrict__ W3, const float* __restrict__ b3,
                       float* __restrict__ out) {
  __shared__ float hp[512], h[32], h2[32], lg[10];
  int b = blockIdx.x, tid = threadIdx.x;
  for (int e = tid; e < 512; e += 128) {
    int k = e >> 5, f = e & 31;
    float acc = bcat[f];
    const float* rg = x2g + (size_t)b * 512 + k * 32;
    const float* rc = x2c + (size_t)b * 512 + k * 32;
    for (int j = 0; j < 32; j++)
      acc += rg[j] * Wcat[j * 32 + f] + rc[j] * Wcat[(32 + j) * 32 + f];
    hp[e] = fmaxf(acc, 0.f);
  }
  __syncthreads();
  if (tid < 32) {
    float su = 0;
    for (int k = 0; k < 16; k++) su += hp[k * 32 + tid];
    h[tid] = su;
  }
  __syncthreads();
  if (tid < 32) {
    float acc = b2[tid];
    for (int j = 0; j < 32; j++) acc += h[j] * W2[j * 32 + tid];
    h2[tid] = fmaxf(acc, 0.f);
  }
  __syncthreads();
  if (tid < 10) {
    float acc = b3[tid];
    for (int j = 0; j < 32; j++) acc += h2[j] * W3[j * 10 + tid];
    lg[tid] = acc;
  }
  __syncthreads();
  if (tid == 0) {
    float mx = -1e30f;
    for (int o = 0; o < 10; o++) mx = fmaxf(mx, lg[o]);
    float se = 0;
    for (int o = 0; o < 10; o++) se += __expf(lg[o] - mx);
    float lse = mx + __logf(se);
    for (int o = 0; o < 10; o++) out[b * 10 + o] = lg[o] - lse;
  }
}

// ------------- K12: loss assembly -------------
__global__ void k_losses(const float* __restrict__ scal, const float* __restrict__ ss_ct,
                         const float* __restrict__ ssG, const float* __restrict__ ssC,
                         const float* __restrict__ numG, const float* __restrict__ denG,
                         const float* __restrict__ numC, const float* __restrict__ denC,
                         float* __restrict__ out) {
  __shared__ float buf[256];
  int tid = threadIdx.x;
  float olct = 0, olg = 0, olc = 0;
  const float r32 = 1.f / sqrtf(32.f);
  const float r16 = 1.f / sqrtf(16.f);
  for (int b = 0; b < Bn; b++) {
    const float* m = ss_ct + b * 1024;
    float ssq = 0;
    for (int e = tid; e < 1024; e += 256) ssq += m[e] * m[e];
    float nrm = sqrtf(block_reduce(ssq, buf));
    float dsq = 0;
    for (int e = tid; e < 1024; e += 256) {
      float v = m[e] / nrm - (((e >> 5) == (e & 31)) ? r32 : 0.f);
      dsq += v * v;
    }
    olct += sqrtf(block_reduce(dsq, buf));
    float vg = ssG[b * 256 + tid];
    float ng = sqrtf(block_reduce(vg * vg, buf));
    float dg = vg / ng - (((tid >> 4) == (tid & 15)) ? r16 : 0.f);
    olg += sqrtf(block_reduce(dg * dg, buf));
    float vc = ssC[b * 256 + tid];
    float nc = sqrtf(block_reduce(vc * vc, buf));
    float dc = vc / nc - (((tid >> 4) == (tid & 15)) ? r16 : 0.f);
    olc += sqrtf(block_reduce(dc * dc, buf));
  }
  if (tid == 0) {
    olct /= Bn; olg /= Bn; olc /= Bn;
    float mc_rw = 0, ctl = 0, mc1 = 0, mc2 = 0, olrw = 0;
    const float r2 = 1.f / sqrtf(2.f);
    for (int b = 0; b < Bn; b++) {
      mc_rw += -(scal[0 * 16 + b] / (scal[1 * 16 + b] + EPSF));
      ctl += (scal[3 * 16 + b] - scal[2 * 16 + b]) / (scal[4 * 16 + b] + EPSF);
      mc1 += -(numG[b] / (denG[b] + EPSF));
      mc2 += -(numC[b] / (denC[b] + EPSF));
      float a = scal[6 * 16 + b], o = scal[7 * 16 + b], d = scal[8 * 16 + b];
      float nrm = sqrtf(a * a + 2.f * o * o + d * d);
      float d00 = a / nrm - r2, d01 = o / nrm, d11 = d / nrm - r2;
      olrw += sqrtf(d00 * d00 + 2.f * d01 * d01 + d11 * d11);
    }
    mc_rw /= Bn; ctl /= Bn; mc1 /= Bn; mc2 /= Bn; olrw /= Bn;
    out[0] = mc_rw + ctl + mc1 + mc2;
    out[1] = olrw + olct + olg + olc;
  }
}

extern "C" void kernel_launch(void* const* d_in, const int* in_sizes, int n_in,
                              void* d_out, int out_size, void* d_ws, size_t ws_size,
                              hipStream_t stream) {
  (void)in_sizes; (void)n_in; (void)out_size; (void)ws_size;
  const float* x = (const float*)d_in[0];
  const float* adj = (const float*)d_in[1];
  const float* W1 = (const float*)d_in[3];     const float* b1 = (const float*)d_in[4];
  const float* Wrw = (const float*)d_in[5];    const float* brw = (const float*)d_in[6];
  const float* Wg_rel = (const float*)d_in[7]; const float* bg = (const float*)d_in[8];
  const float* Wg_root = (const float*)d_in[9];
  const float* Wmcg = (const float*)d_in[10];  const float* bmcg = (const float*)d_in[11];
  const float* W2g_rel = (const float*)d_in[12]; const float* b2g = (const float*)d_in[13];
  const float* W2g_root = (const float*)d_in[14];
  const float* Wct = (const float*)d_in[15];   const float* bct = (const float*)d_in[16];
  const float* Wc_rel = (const float*)d_in[17]; const float* bc = (const float*)d_in[18];
  const float* Wc_root = (const float*)d_in[19];
  const float* Wmcc = (const float*)d_in[20];  const float* bmcc = (const float*)d_in[21];
  const float* Wcat = (const float*)d_in[22];  const float* bcat = (const float*)d_in[23];
  const float* W2 = (const float*)d_in[24];    const float* b2 = (const float*)d_in[25];
  const float* W3 = (const float*)d_in[26];    const float* b3 = (const float*)d_in[27];
  float* out = (float*)d_out;

  float* w = (float*)d_ws;
  size_t off = 0;
  auto alloc = [&](size_t n) { float* p = w + off; off += n; return p; };
  float* x1 = alloc((size_t)Bn * Nn * Hf);
  float* Bc = alloc((size_t)Bn * Nn * 48);
  float* sqv = alloc((size_t)Bn * Nn);
  float* Y1 = alloc((size_t)Bn * Nn * 48);
  float* scal = alloc(16 * 16);
  float* ssct = alloc((size_t)Bn * 1024);
  float* AXg = alloc((size_t)Bn * Nn * Hf);
  float* AXc = alloc((size_t)Bn * Nn * Hf);
  float* xg = alloc((size_t)Bn * Nn * Hf);
  float* xc = alloc((size_t)Bn * Nn * Hf);
  float* smg = alloc((size_t)Bn * Nn * KMC);
  float* smc = alloc((size_t)Bn * Nn * KMC);
  float* ASg = alloc((size_t)Bn * Nn * KMC);
  float* ASc = alloc((size_t)Bn * Nn * KMC);
  float* dgr = alloc((size_t)Bn * Nn);
  float* dcr = alloc((size_t)Bn * Nn);
  float* oaG = alloc(Bn * 256); float* oaC = alloc(Bn * 256);
  float* ssG = alloc(Bn * 256); float* ssC = alloc(Bn * 256);
  float* ftG = alloc(Bn * 512); float* ftC = alloc(Bn * 512);
  float* denG = alloc(16); float* denC = alloc(16);
  float* numG = alloc(16); float* numC = alloc(16);
  float* x2g = alloc(Bn * 512); float* x2c = alloc(Bn * 512);
  float* volp = scal + 5 * 16;

  int nodes = Bn * Nn;
  k_lin1<<<Bn * NT, 32, 0, stream>>>(x, W1, b1, x1);
  k_assign<<<(nodes + 255) / 256, 256, 0, stream>>>(x1, Wrw, brw, Wct, bct, Bc, sqv);
  k_pass1<<<Bn * NT, 32, 0, stream>>>(adj, Bc, Y1);
  k_reduce1<<<Bn, 256, 0, stream>>>(Bc, Y1, sqv, scal);
  k_ssct<<<Bn, 256, 0, stream>>>(Bc, ssct);
  k_pass2<<<Bn * NT, 32, 0, stream>>>(adj, Bc, sqv, x1, volp, AXg, AXc);
  k_dgc<<<(nodes * Hf + 255) / 256, 256, 0, stream>>>(AXg, x1, Wg_rel, bg, Wg_root, xg);
  k_dgc<<<(nodes * Hf + 255) / 256, 256, 0, stream>>>(AXc, x1, Wc_rel, bc, Wc_root, xc);
  k_softmax16<<<(nodes + 255) / 256, 256, 0, stream>>>(xg, Wmcg, bmcg, smg);
  k_softmax16<<<(nodes + 255) / 256, 256, 0, stream>>>(xc, Wmcc, bmcc, smc);
  k_pass3<<<Bn * NT, 32, 0, stream>>>(adj, Bc, sqv, smg, smc, volp, ASg, ASc, dgr, dcr);
  k_pool_reduce<<<Bn, 256, 0, stream>>>(smg, ASg, xg, dgr, oaG, ssG, ftG, denG);
  k_pool_reduce<<<Bn, 256, 0, stream>>>(smc, ASc, xc, dcr, oaC, ssC, ftC, denC);
  k_pool_fin<<<Bn, 256, 0, stream>>>(oaG, ftG, W2g_rel, b2g, W2g_root, x2g, numG);
  k_pool_fin<<<Bn, 256, 0, stream>>>(oaC, ftC, W2g_rel, b2g, W2g_root, x2c, numC);
  k_head<<<Bn, 128, 0, stream>>>(x2g, x2c, Wcat, bcat, W2, b2, W3, b3, out);
  k_losses<<<1, 256, 0, stream>>>(scal, ssct, ssG, ssC, numG, denG, numC, denC, out + 160);
}